// HalfKPInputLayer_16003048145507
// MI455X (gfx1250) — compile-verified
//
#include <hip/hip_runtime.h>

// NNUE input layer: out[0:256]   = relu(W_my  @ x[0:41024]     + b_my)
//                   out[256:512] = relu(W_opp @ x[41024:82048] + b_opp)
//
// Bandwidth-bound GEMV (84 MB of weights, 42 MFLOP). Strategy:
//  - f32 WMMA (V_WMMA_F32_16X16X4_F32), 16-row tiles, K split into 512-wide
//    chunks across 2592 single-wave blocks for chip-wide HBM/L2 streaming.
//  - partial sums combined with global_atomic_add_f32 into a bias-seeded
//    accumulator in d_ws; separate ReLU finalize kernel.

#define FEAT 41024
#define HIDDEN 256
#define KCHUNK 512

typedef __attribute__((ext_vector_type(2))) float v2f;
typedef __attribute__((ext_vector_type(8))) float v8f;

__global__ __launch_bounds__(64) void nnue_init_accum(const float* __restrict__ b_my,
                                                      const float* __restrict__ b_opp,
                                                      float* __restrict__ accum) {
    int i = blockIdx.x * blockDim.x + threadIdx.x;
    if (i < HIDDEN)          accum[i] = b_my[i];
    else if (i < 2 * HIDDEN) accum[i] = b_opp[i - HIDDEN];
}

__global__ __launch_bounds__(32) void nnue_gemv_wmma(const float* __restrict__ x,
                                                     const float* __restrict__ Wmy,
                                                     const float* __restrict__ Wopp,
                                                     float* __restrict__ accum) {
    const int lane = threadIdx.x;            // 0..31, one wave per block
    const int mat  = blockIdx.z;             // 0 = my, 1 = opp
    const int tile = blockIdx.y;             // 16-row tile of the 256 outputs
    const int kb0  = blockIdx.x * KCHUNK;
    const int kend = min(kb0 + KCHUNK, FEAT);   // FEAT % 4 == 0, chunks stay /4

    const float* __restrict__ W  = mat ? Wopp : Wmy;
    const float* __restrict__ xs = x + mat * FEAT;

    // A fragment (16x4 f32): lane m = lane&15 holds row tile*16+m;
    // VGPR0/1 = K={0,1} for lanes 0-15, K={2,3} for lanes 16-31.
    const int m     = lane & 15;
    const int khalf = (lane >> 4) << 1;      // 0 or 2
    const float* __restrict__ wrow = W + (size_t)(tile * 16 + m) * FEAT + khalf;
    const float* __restrict__ xp   = xs + khalf;

    v8f acc = {};
    #pragma unroll 8
    for (int kb = kb0; kb < kend; kb += 4) {
        v2f a = *(const v2f*)(wrow + kb);    // global_load_b64, 8B aligned
        v2f xv = *(const v2f*)(xp + kb);     // tiny x stream, L2-hot
        // B fragment (4x16 f32): only column n = lane%16 == 0 carries x,
        // K split mirrors A (k = vgpr + 2*lanehalf); other columns zero.
        v2f b;
        b.x = (m == 0) ? xv.x : 0.0f;
        b.y = (m == 0) ? xv.y : 0.0f;
        acc = __builtin_amdgcn_wmma_f32_16x16x4_f32(
            /*neg_a=*/false, a, /*neg_b=*/false, b,
            /*c_mod=*/(short)0, acc, /*reuse_a=*/false, /*reuse_b=*/false);
    }

    // D (16x16 f32): element (M,N) at lane N + 16*(M/8), VGPR M%8.
    // Column 0 lives in lanes 0 (M=0..7) and 16 (M=8..15).
    if (m == 0) {
        float* dst = accum + mat * HIDDEN + tile * 16 + (lane >> 4) * 8;
        #pragma unroll
        for (int v = 0; v < 8; ++v)
            atomicAdd(dst + v, acc[v]);      // global_atomic_add_f32
    }
}

__global__ __launch_bounds__(64) void nnue_relu_out(const float* __restrict__ accum,
                                                    float* __restrict__ out) {
    int i = blockIdx.x * blockDim.x + threadIdx.x;
    if (i < 2 * HIDDEN) out[i] = fmaxf(accum[i], 0.0f);
}

extern "C" void kernel_launch(void* const* d_in, const int* in_sizes, int n_in,
                              void* d_out, int out_size, void* d_ws, size_t ws_size,
                              hipStream_t stream) {
    const float* x     = (const float*)d_in[0];   // [2*41024]
    const float* W_my  = (const float*)d_in[1];   // [256, 41024]
    const float* b_my  = (const float*)d_in[2];   // [256]
    const float* W_opp = (const float*)d_in[3];   // [256, 41024]
    const float* b_opp = (const float*)d_in[4];   // [256]
    float* out   = (float*)d_out;                 // [512]
    float* accum = (float*)d_ws;                  // 512 floats of scratch

    nnue_init_accum<<<dim3((2 * HIDDEN + 63) / 64), 64, 0, stream>>>(b_my, b_opp, accum);

    const int nchunks = (FEAT + KCHUNK - 1) / KCHUNK;   // 81
    nnue_gemv_wmma<<<dim3(nchunks, HIDDEN / 16, 2), 32, 0, stream>>>(x, W_my, W_opp, accum);

    nnue_relu_out<<<dim3((2 * HIDDEN + 63) / 64), 64, 0, stream>>>(accum, out);
}